// Net_89378269430261
// MI455X (gfx1250) — compile-verified
//
#include <hip/hip_runtime.h>
#include <cstddef>

typedef __attribute__((ext_vector_type(2))) float v2f;
typedef __attribute__((ext_vector_type(8))) float v8f;

#define EPSV 1e-5f

// ---------------- utility ----------------
__global__ void k_fill(float* __restrict__ p, float v, long n) {
  long i = (long)blockIdx.x * blockDim.x + threadIdx.x;
  long stride = (long)gridDim.x * blockDim.x;
  for (; i < n; i += stride) p[i] = v;
}

// degree accumulation over edge rows (self-loop handled by init=1)
__global__ void k_deg_accum(const int* __restrict__ ei, float* __restrict__ deg, int E) {
  int i = blockIdx.x * blockDim.x + threadIdx.x;
  int stride = gridDim.x * blockDim.x;
  for (; i < E; i += stride) atomicAdd(&deg[ei[i]], 1.0f);
}

__global__ void k_dis(const float* __restrict__ deg, float* __restrict__ dis, int N) {
  int i = blockIdx.x * blockDim.x + threadIdx.x;
  if (i < N) dis[i] = rsqrtf(deg[i]);
}

// ---------------- GEMM1: h1 = relu(BN(x @ W1 + b1))  [N,128]x[128,256] ----------------
__global__ __launch_bounds__(256)
void k_gemm1(const float* __restrict__ x, const float* __restrict__ W1,
             const float* __restrict__ b1, const float* __restrict__ g1,
             const float* __restrict__ be1, const float* __restrict__ m1,
             const float* __restrict__ v1, float* __restrict__ h1) {
  __shared__ float As[16 * 132];                 // +4 pad -> conflict-free column reads
  const int m0 = blockIdx.x * 16;
  const int tid = threadIdx.x;
  for (int i = tid; i < 16 * 128; i += 256) {    // rows m0..m0+15 are contiguous in x
    int r = i >> 7, c = i & 127;
    As[r * 132 + c] = x[(long)(m0 + r) * 128 + c];
  }
  __syncthreads();
  const int wave = tid >> 5, lane = tid & 31;
  const int nl = lane & 15, kh = lane >> 4;      // kh: which K-half this lane holds
  const int n0 = (blockIdx.y * 8 + wave) * 16;
  v8f c = {};
  for (int k0 = 0; k0 < 128; k0 += 4) {
    v2f a, b;
    a.x = As[nl * 132 + k0 + 2 * kh];
    a.y = As[nl * 132 + k0 + 2 * kh + 1];
    b.x = W1[(long)(k0 + 2 * kh) * 256 + n0 + nl];
    b.y = W1[(long)(k0 + 2 * kh + 1) * 256 + n0 + nl];
    c = __builtin_amdgcn_wmma_f32_16x16x4_f32(false, a, false, b, (short)0, c, false, false);
  }
  const int n = n0 + nl;
  const float s = rsqrtf(v1[n] + EPSV) * g1[n];
  const float bias = b1[n], mu = m1[n], beta = be1[n];
#pragma unroll
  for (int r = 0; r < 8; ++r) {
    int row = m0 + r + 8 * kh;
    float val = (c[r] + bias - mu) * s + beta;
    h1[(long)row * 256 + n] = fmaxf(val, 0.0f);
  }
}

// ---------------- GEMM2: h2 = h1 @ W2 + b2  [N,256]x[256,64] ----------------
__global__ __launch_bounds__(128)
void k_gemm2(const float* __restrict__ h1, const float* __restrict__ W2,
             const float* __restrict__ b2, float* __restrict__ h2) {
  __shared__ float As[16 * 260];
  const int m0 = blockIdx.x * 16;
  const int tid = threadIdx.x;
  for (int i = tid; i < 16 * 256; i += 128) {
    int r = i >> 8, cc = i & 255;
    As[r * 260 + cc] = h1[(long)(m0 + r) * 256 + cc];
  }
  __syncthreads();
  const int wave = tid >> 5, lane = tid & 31;
  const int nl = lane & 15, kh = lane >> 4;
  const int n0 = wave * 16;
  v8f c = {};
  for (int k0 = 0; k0 < 256; k0 += 4) {
    v2f a, b;
    a.x = As[nl * 260 + k0 + 2 * kh];
    a.y = As[nl * 260 + k0 + 2 * kh + 1];
    b.x = W2[(long)(k0 + 2 * kh) * 64 + n0 + nl];
    b.y = W2[(long)(k0 + 2 * kh + 1) * 64 + n0 + nl];
    c = __builtin_amdgcn_wmma_f32_16x16x4_f32(false, a, false, b, (short)0, c, false, false);
  }
  const int n = n0 + nl;
  const float bias = b2[n];
#pragma unroll
  for (int r = 0; r < 8; ++r) {
    int row = m0 + r + 8 * kh;
    h2[(long)row * 64 + n] = c[r] + bias;
  }
}

// ---------------- message passing: agg[col] += norm*ea * h2[row] (wave per edge) ----------------
__global__ void k_msg(const int* __restrict__ ei, const float* __restrict__ ea,
                      const float* __restrict__ dis, const float* __restrict__ h2,
                      float* __restrict__ agg, int E) {
  const int lane = threadIdx.x & 31;
  int wave = (blockIdx.x * blockDim.x + threadIdx.x) >> 5;
  const int nw = (gridDim.x * blockDim.x) >> 5;
  for (int e = wave; e < E; e += nw) {
    int r = ei[e], c = ei[E + e];                 // wave-uniform -> scalarized
    float w = dis[r] * dis[c] * ea[e];
    float2 v = ((const float2*)(h2 + (long)r * 64))[lane];
    int en = e + nw;                              // prefetch next edge's source row
    if (en < E) __builtin_prefetch(h2 + (long)ei[en] * 64 + lane * 2, 0, 1);
    float* dst = agg + (long)c * 64 + lane * 2;
    atomicAdd(dst, w * v.x);
    atomicAdd(dst + 1, w * v.y);
  }
}

// ---------------- relu + self-loop + global mean pool accumulation ----------------
__global__ void k_relu_pool(const float* __restrict__ agg, const float* __restrict__ h2,
                            const float* __restrict__ dis, const int* __restrict__ batch,
                            float* __restrict__ gsum, float* __restrict__ cnt, int N) {
  const int lane = threadIdx.x & 31;
  int wave = (blockIdx.x * blockDim.x + threadIdx.x) >> 5;
  const int nw = (gridDim.x * blockDim.x) >> 5;
  for (int i = wave; i < N; i += nw) {
    int g = batch[i];
    float w = dis[i] * dis[i];                    // self-loop norm = 1/deg
    float2 a = ((const float2*)(agg + (long)i * 64))[lane];
    float2 h = ((const float2*)(h2 + (long)i * 64))[lane];
    float v0 = fmaxf(a.x + w * h.x, 0.0f);
    float v1 = fmaxf(a.y + w * h.y, 0.0f);
    float* dst = gsum + g * 64 + lane * 2;
    atomicAdd(dst, v0);
    atomicAdd(dst + 1, v1);
    if (lane == 0) atomicAdd(&cnt[g], 1.0f);
  }
}

__global__ void k_pooled(const float* __restrict__ gsum, const float* __restrict__ cnt,
                         float* __restrict__ pooled) {
  int g = blockIdx.x, d = threadIdx.x;
  pooled[g * 64 + d] = gsum[g * 64 + d] / fmaxf(cnt[g], 1.0f);
}

// ---------------- head GEMM: t = relu(BN(pooled @ W3 + b3))  64x64x64, 16 tiles = 16 waves ----------------
__global__ __launch_bounds__(512)
void k_head1(const float* __restrict__ pooled, const float* __restrict__ W3,
             const float* __restrict__ b3, const float* __restrict__ g2,
             const float* __restrict__ be2, const float* __restrict__ m2,
             const float* __restrict__ v2, float* __restrict__ t) {
  const int tid = threadIdx.x;
  const int wave = tid >> 5, lane = tid & 31;
  const int nl = lane & 15, kh = lane >> 4;
  const int m0 = (wave >> 2) * 16, n0 = (wave & 3) * 16;
  v8f c = {};
  for (int k0 = 0; k0 < 64; k0 += 4) {
    v2f a, b;
    a.x = pooled[(m0 + nl) * 64 + k0 + 2 * kh];
    a.y = pooled[(m0 + nl) * 64 + k0 + 2 * kh + 1];
    b.x = W3[(k0 + 2 * kh) * 64 + n0 + nl];
    b.y = W3[(k0 + 2 * kh + 1) * 64 + n0 + nl];
    c = __builtin_amdgcn_wmma_f32_16x16x4_f32(false, a, false, b, (short)0, c, false, false);
  }
  const int n = n0 + nl;
  const float s = rsqrtf(v2[n] + EPSV) * g2[n];
  const float bias = b3[n], mu = m2[n], beta = be2[n];
#pragma unroll
  for (int r = 0; r < 8; ++r) {
    int row = m0 + r + 8 * kh;
    float val = (c[r] + bias - mu) * s + beta;
    t[row * 64 + n] = fmaxf(val, 0.0f);
  }
}

__global__ void k_head2(const float* __restrict__ t, const float* __restrict__ W4,
                        const float* __restrict__ b4, float* __restrict__ out, int G) {
  int g = blockIdx.x * blockDim.x + threadIdx.x;
  if (g < G) {
    float s = b4[0];
    for (int d = 0; d < 64; ++d) s += t[g * 64 + d] * W4[d];
    out[g] = s;
  }
}

// ---------------- launch ----------------
extern "C" void kernel_launch(void* const* d_in, const int* in_sizes, int n_in,
                              void* d_out, int out_size, void* d_ws, size_t ws_size,
                              hipStream_t stream) {
  const float* x   = (const float*)d_in[0];
  const float* ea  = (const float*)d_in[1];
  const float* W1  = (const float*)d_in[2];
  const float* b1  = (const float*)d_in[3];
  const float* g1  = (const float*)d_in[4];
  const float* be1 = (const float*)d_in[5];
  const float* m1  = (const float*)d_in[6];
  const float* v1  = (const float*)d_in[7];
  const float* W2  = (const float*)d_in[8];
  const float* b2  = (const float*)d_in[9];
  const float* W3  = (const float*)d_in[10];
  const float* b3  = (const float*)d_in[11];
  const float* g2  = (const float*)d_in[12];
  const float* be2 = (const float*)d_in[13];
  const float* m2  = (const float*)d_in[14];
  const float* v2  = (const float*)d_in[15];
  const float* W4  = (const float*)d_in[16];
  const float* b4  = (const float*)d_in[17];
  const int* ei    = (const int*)d_in[18];
  const int* batch = (const int*)d_in[19];
  float* out = (float*)d_out;

  const int N = in_sizes[0] / 128;   // 50000 (multiple of 16)
  const int E = in_sizes[1];         // 800000
  const int G = 64;

  float* ws     = (float*)d_ws;
  float* h1     = ws;                       // N*256
  float* h2     = h1 + (size_t)N * 256;     // N*64
  float* deg    = h2 + (size_t)N * 64;      // N
  float* dis    = deg + N;                  // N
  float* agg    = dis + N;                  // N*64
  float* gsum   = agg + (size_t)N * 64;     // G*64
  float* cnt    = gsum + (size_t)G * 64;    // G (contiguous after gsum)
  float* pooled = cnt + G;                  // G*64
  float* t      = pooled + (size_t)G * 64;  // G*64

  // degrees (self loops contribute 1)
  k_fill<<<512, 256, 0, stream>>>(deg, 1.0f, (long)N);
  k_deg_accum<<<1024, 256, 0, stream>>>(ei, deg, E);
  k_dis<<<(N + 255) / 256, 256, 0, stream>>>(deg, dis, N);

  // node MLP (WMMA f32)
  k_gemm1<<<dim3(N / 16, 2), 256, 0, stream>>>(x, W1, b1, g1, be1, m1, v1, h1);
  k_gemm2<<<N / 16, 128, 0, stream>>>(h1, W2, b2, h2);

  // message passing
  k_fill<<<2048, 256, 0, stream>>>(agg, 0.0f, (long)N * 64);
  k_msg<<<4096, 256, 0, stream>>>(ei, ea, dis, h2, agg, E);

  // pooling (gsum and cnt are contiguous: one zero-fill)
  k_fill<<<4, 256, 0, stream>>>(gsum, 0.0f, (long)(G * 64 + G));
  k_relu_pool<<<1024, 256, 0, stream>>>(agg, h2, dis, batch, gsum, cnt, N);
  k_pooled<<<G, 64, 0, stream>>>(gsum, cnt, pooled);

  // head MLP
  k_head1<<<1, 512, 0, stream>>>(pooled, W3, b3, g2, be2, m2, v2, t);
  k_head2<<<1, 64, 0, stream>>>(t, W4, b4, out, G);
}